// ECGFeatureExtractor_mamba_1236950581279
// MI455X (gfx1250) — compile-verified
//
#include <hip/hip_runtime.h>
#include <hip/hip_bf16.h>
#include <math.h>

// ---------------------------------------------------------------------------
// ECG feature extractor + 2 Mamba blocks for MI455X (gfx1250, wave32).
// All dense projections run through V_WMMA_F32_16X16X4_F32 (exact fp32 FMA
// on the matrix pipe). Convs / scan are VALU kernels (bandwidth bound).
// Workspace requirement: 128 MB (reuses regions; see offsets below).
// ---------------------------------------------------------------------------

typedef float v2f __attribute__((ext_vector_type(2)));
typedef float v8f __attribute__((ext_vector_type(8)));

#define BATCH 512
#define L_IN  4096
#define GEMM_WAVES 4

__device__ __forceinline__ float softplus_f(float x) {
  // stable log(1+exp(x))
  return (x > 0.f) ? (x + log1pf(expf(-x))) : log1pf(expf(x));
}
__device__ __forceinline__ float silu_f(float x) {
  return x / (1.f + expf(-x));
}

// ---------------- Stage 1: conv1d(1->16,k5,p2) + relu + maxpool2 -----------
__global__ void k_conv1_pool(const float* __restrict__ x, const float* __restrict__ w,
                             const float* __restrict__ bias, float* __restrict__ out) {
  int idx = blockIdx.x * blockDim.x + threadIdx.x;   // B*16*2048
  int lo = idx & 2047;
  int c  = (idx >> 11) & 15;
  int b  = idx >> 15;
  const float* xb = x + (size_t)b * L_IN;
  float wv[5];
#pragma unroll
  for (int k = 0; k < 5; ++k) wv[k] = w[c * 5 + k];
  float bv = bias[c];
  float m = -1e30f;
#pragma unroll
  for (int p = 0; p < 2; ++p) {
    int t0 = 2 * lo + p - 2;
    float s = bv;
#pragma unroll
    for (int k = 0; k < 5; ++k) {
      int t = t0 + k;
      if (t >= 0 && t < L_IN) s += wv[k] * xb[t];
    }
    m = fmaxf(m, s);
  }
  out[idx] = fmaxf(m, 0.f);   // relu(max) == max(relu)
}

// ---------------- Stage 2: conv1d(16->32,k5,p2) + relu + maxpool2 ----------
__global__ void k_conv2_pool(const float* __restrict__ in, const float* __restrict__ w,
                             const float* __restrict__ bias, float* __restrict__ out) {
  __shared__ float sw[32 * 16 * 5];
  for (int i = threadIdx.x; i < 32 * 16 * 5; i += blockDim.x) sw[i] = w[i];
  __syncthreads();
  int idx = blockIdx.x * blockDim.x + threadIdx.x;   // B*32*1024
  int lo = idx & 1023;
  int c  = (idx >> 10) & 31;
  int b  = idx >> 15;
  const float* ib = in + (size_t)b * 16 * 2048;
  float m = -1e30f;
#pragma unroll
  for (int p = 0; p < 2; ++p) {
    int t0 = 2 * lo + p - 2;
    float s = bias[c];
    for (int ic = 0; ic < 16; ++ic) {
      const float* row = ib + ic * 2048;
      const float* wr  = sw + (c * 16 + ic) * 5;
#pragma unroll
      for (int k = 0; k < 5; ++k) {
        int t = t0 + k;
        if (t >= 0 && t < 2048) s += wr[k] * row[t];
      }
    }
    m = fmaxf(m, s);
  }
  out[idx] = fmaxf(m, 0.f);
}

// -------- Stage 3: conv1d(32->64,k5,p2) + relu fused with avgpool(32) ------
__global__ void k_conv3_pool(const float* __restrict__ in, const float* __restrict__ w,
                             const float* __restrict__ bias, float* __restrict__ out) {
  __shared__ float sw[64 * 32 * 5];   // 40 KB LDS
  for (int i = threadIdx.x; i < 64 * 32 * 5; i += blockDim.x) sw[i] = w[i];
  __syncthreads();
  int idx = blockIdx.x * blockDim.x + threadIdx.x;   // B*64*32
  int j = idx & 31;
  int c = (idx >> 5) & 63;
  int b = idx >> 11;
  const float* ib = in + (size_t)b * 32 * 1024;
  float bv = bias[c];
  float sum = 0.f;
  for (int t = 0; t < 32; ++t) {
    int p0 = j * 32 + t - 2;
    float s = bv;
    for (int ic = 0; ic < 32; ++ic) {
      const float* row = ib + ic * 1024;
      const float* wr  = sw + (c * 32 + ic) * 5;
#pragma unroll
      for (int k = 0; k < 5; ++k) {
        int p = p0 + k;
        if (p >= 0 && p < 1024) s += wr[k] * row[p];
      }
    }
    sum += fmaxf(s, 0.f);
  }
  // flatten layout [B, 64*32]: col = c*32 + j
  out[(size_t)b * 2048 + c * 32 + j] = sum * (1.f / 32.f);
}

// ---------------- Generic WMMA fp32 NT GEMM: D = A * W^T (+bias)(+act) -----
// A: [M,K] row-major (row stride lda); W: [N,K] row-major (row stride ldw)
// D: [M,N] (row stride ldc). One 16x16 output tile per wave, K stepped by 4
// through V_WMMA_F32_16X16X4_F32. M must be a multiple of 16; N arbitrary.
__global__ void k_wmma_gemm_nt(const float* __restrict__ A, const float* __restrict__ W,
                               const float* __restrict__ bias, float* __restrict__ D,
                               int N, int K, int lda, int ldw, int ldc, int act) {
  int lane = threadIdx.x & 31;
  int wave = threadIdx.x >> 5;
  int n0 = (blockIdx.x * GEMM_WAVES + wave) * 16;
  int m0 = blockIdx.y * 16;
  if (n0 >= N) return;                       // wave-uniform exit

  int mi = lane & 15;
  int kh = (lane >> 4) << 1;                 // lanes 0-15: K=0,1  lanes 16-31: K=2,3
  const float* ap = A + (size_t)(m0 + mi) * lda + kh;
  int nc = n0 + mi;
  bool nvalid = nc < N;
  // clamp B row for tail lanes: their output column is never stored, and a
  // WMMA output column depends only on its own B column, so garbage is safe.
  const float* wp = W + (size_t)(nvalid ? nc : 0) * ldw + kh;

  v8f acc = {0.f, 0.f, 0.f, 0.f, 0.f, 0.f, 0.f, 0.f};
  for (int k = 0; k < K; k += 4) {
    v2f av = *(const v2f*)(ap + k);
    v2f bv = *(const v2f*)(wp + k);
    acc = __builtin_amdgcn_wmma_f32_16x16x4_f32(false, av, false, bv,
                                                (short)0, acc, false, false);
  }
  if (nvalid) {
    float bval = bias ? bias[nc] : 0.f;
    int mbase = m0 + ((lane >> 4) << 3);     // +8 rows for upper half-wave
#pragma unroll
    for (int r = 0; r < 8; ++r) {
      float v = acc[r] + bval;
      if (act == 1) v = softplus_f(v);
      D[(size_t)(mbase + r) * ldc + nc] = v;
    }
  }
}

// ---------------- LayerNorm over last dim (64), one wave per row -----------
__global__ void k_layernorm(const float* __restrict__ in, const float* __restrict__ g,
                            const float* __restrict__ bta, float* __restrict__ out,
                            int rows) {
  int row = blockIdx.x * (blockDim.x >> 5) + (threadIdx.x >> 5);
  int lane = threadIdx.x & 31;
  if (row >= rows) return;
  const float* p = in + (size_t)row * 64;
  float x0 = p[lane], x1 = p[lane + 32];
  float s = x0 + x1;
#pragma unroll
  for (int off = 16; off > 0; off >>= 1) s += __shfl_xor(s, off, 32);
  float mean = s * (1.f / 64.f);
  float d0 = x0 - mean, d1 = x1 - mean;
  float v = d0 * d0 + d1 * d1;
#pragma unroll
  for (int off = 16; off > 0; off >>= 1) v += __shfl_xor(v, off, 32);
  float inv = 1.f / sqrtf(v * (1.f / 64.f) + 1e-5f);
  float* q = out + (size_t)row * 64;
  q[lane]      = d0 * inv * g[lane]      + bta[lane];
  q[lane + 32] = d1 * inv * g[lane + 32] + bta[lane + 32];
}

// -------- Causal depthwise conv (k=2) + SiLU on xi half of xz --------------
__global__ void k_dwconv_silu(const float* __restrict__ xz, const float* __restrict__ cw,
                              const float* __restrict__ cb, float* __restrict__ u,
                              int total) {
  int idx = blockIdx.x * blockDim.x + threadIdx.x;   // B*L*64
  if (idx >= total) return;
  int d  = idx & 63;
  int bl = idx >> 6;
  int l  = bl & 63;
  float xc = xz[(size_t)bl * 128 + d];
  float xp = (l > 0) ? xz[(size_t)(bl - 1) * 128 + d] : 0.f;
  float v = cw[d * 2] * xp + cw[d * 2 + 1] * xc + cb[d];
  u[idx] = silu_f(v);
}

// -------- Selective scan: thread per (batch, channel), 4 states ------------
__global__ void k_scan(const float* __restrict__ dt, const float* __restrict__ xdbl,
                       const float* __restrict__ u, const float* __restrict__ xz,
                       const float* __restrict__ alog, const float* __restrict__ Dp,
                       float* __restrict__ y) {
  int idx = blockIdx.x * blockDim.x + threadIdx.x;   // B*64
  if (idx >= BATCH * 64) return;
  int d = idx & 63;
  int b = idx >> 6;
  float A0 = -expf(alog[d * 4 + 0]);
  float A1 = -expf(alog[d * 4 + 1]);
  float A2 = -expf(alog[d * 4 + 2]);
  float A3 = -expf(alog[d * 4 + 3]);
  float Dd = Dp[d];
  float h0 = 0.f, h1 = 0.f, h2 = 0.f, h3 = 0.f;
  for (int l = 0; l < 64; ++l) {
    size_t bl = (size_t)b * 64 + l;
    float dtv = dt[bl * 64 + d];
    float uv  = u[bl * 64 + d];
    const float* xd = xdbl + bl * 12;   // [dt(4) | B(4) | C(4)]
    float dbu = dtv * uv;
    h0 = expf(dtv * A0) * h0 + dbu * xd[4];
    h1 = expf(dtv * A1) * h1 + dbu * xd[5];
    h2 = expf(dtv * A2) * h2 + dbu * xd[6];
    h3 = expf(dtv * A3) * h3 + dbu * xd[7];
    float yv = h0 * xd[8] + h1 * xd[9] + h2 * xd[10] + h3 * xd[11] + Dd * uv;
    float z = xz[bl * 128 + 64 + d];
    y[bl * 64 + d] = yv * silu_f(z);
  }
}

// ---------------------------------------------------------------------------
// Host side
// ---------------------------------------------------------------------------
static inline void run_gemm(const float* A, const float* W, const float* bias, float* D,
                            int M, int N, int K, int lda, int ldw, int ldc, int act,
                            hipStream_t s) {
  int ntiles = (N + 15) / 16;
  dim3 grid((ntiles + GEMM_WAVES - 1) / GEMM_WAVES, M / 16);
  dim3 block(GEMM_WAVES * 32);
  k_wmma_gemm_nt<<<grid, block, 0, s>>>(A, W, bias, D, N, K, lda, ldw, ldc, act);
}

struct MambaW {
  const float *lng, *lnb, *inw, *cvw, *cvb, *xpw, *dtw, *dtb, *alog, *dp, *ow;
};

static void run_mamba(const float* tokens, const MambaW& w, float* out,
                      float* ln, float* xz, float* u, float* xd, float* dt, float* y,
                      hipStream_t s) {
  const int rows = BATCH * 64;                 // 32768 tokens
  k_layernorm<<<rows / 8, 256, 0, s>>>(tokens, w.lng, w.lnb, ln, rows);
  run_gemm(ln, w.inw, nullptr, xz, rows, 128, 64, 64, 64, 128, 0, s);   // in_proj
  int tot = rows * 64;
  k_dwconv_silu<<<tot / 256, 256, 0, s>>>(xz, w.cvw, w.cvb, u, tot);
  run_gemm(u, w.xpw, nullptr, xd, rows, 12, 64, 64, 64, 12, 0, s);      // x_proj
  run_gemm(xd, w.dtw, w.dtb, dt, rows, 64, 4, 12, 4, 64, 1, s);         // dt + softplus
  k_scan<<<(BATCH * 64) / 256, 256, 0, s>>>(dt, xd, u, xz, w.alog, w.dp, y);
  run_gemm(y, w.ow, nullptr, out, rows, 64, 64, 64, 64, 64, 0, s);      // out_proj
}

extern "C" void kernel_launch(void* const* d_in, const int* in_sizes, int n_in,
                              void* d_out, int out_size, void* d_ws, size_t ws_size,
                              hipStream_t stream) {
  (void)in_sizes; (void)n_in; (void)out_size; (void)ws_size;
  const float* x   = (const float*)d_in[0];
  const float* c1w = (const float*)d_in[1];
  const float* c1b = (const float*)d_in[2];
  const float* c2w = (const float*)d_in[3];
  const float* c2b = (const float*)d_in[4];
  const float* c3w = (const float*)d_in[5];
  const float* c3b = (const float*)d_in[6];
  const float* lw  = (const float*)d_in[7];
  const float* lb  = (const float*)d_in[8];
  MambaW w1 = { (const float*)d_in[9],  (const float*)d_in[10], (const float*)d_in[11],
                (const float*)d_in[12], (const float*)d_in[13], (const float*)d_in[14],
                (const float*)d_in[15], (const float*)d_in[16], (const float*)d_in[17],
                (const float*)d_in[18], (const float*)d_in[19] };
  MambaW w2 = { (const float*)d_in[20], (const float*)d_in[21], (const float*)d_in[22],
                (const float*)d_in[23], (const float*)d_in[24], (const float*)d_in[25],
                (const float*)d_in[26], (const float*)d_in[27], (const float*)d_in[28],
                (const float*)d_in[29], (const float*)d_in[30] };

  // Workspace layout (bytes). Regions reused once their producers are dead.
  char* ws = (char*)d_ws;
  const size_t H2 = 67108864ull;                       // 64 MB
  float* h1     = (float*)(ws);                        // [512,16,2048]  64 MB
  float* h2     = (float*)(ws + H2);                   // [512,32,1024]  64 MB
  float* pooled = (float*)(ws);                        // [512,2048]   (reuses h1)
  float* tok    = (float*)(ws + 8388608ull);           // [512,64,64]  (in h1 region)
  float* mln = (float*)(ws + H2);                      // mamba scratch (reuses h2)
  float* mxz = (float*)(ws + H2 + 8388608ull);
  float* mu  = (float*)(ws + H2 + 25165824ull);
  float* mxd = (float*)(ws + H2 + 33554432ull);
  float* mdt = (float*)(ws + H2 + 35651584ull);
  float* my  = (float*)(ws + H2 + 44040192ull);
  float* t2  = (float*)(ws + H2 + 52428800ull);        // block1 output tokens

  // Conv front-end
  k_conv1_pool<<<(BATCH * 16 * 2048) / 256, 256, 0, stream>>>(x, c1w, c1b, h1);
  k_conv2_pool<<<(BATCH * 32 * 1024) / 256, 256, 0, stream>>>(h1, c2w, c2b, h2);
  k_conv3_pool<<<(BATCH * 64 * 32) / 256, 256, 0, stream>>>(h2, c3w, c3b, pooled);

  // Dense: [512,2048] x [4096,2048]^T + lb -> [512,4096] == [512,64,64]
  run_gemm(pooled, lw, lb, tok, BATCH, 4096, 2048, 2048, 2048, 4096, 0, stream);

  // Mamba blocks
  run_mamba(tok, w1, t2, mln, mxz, mu, mxd, mdt, my, stream);
  run_mamba(t2, w2, (float*)d_out, mln, mxz, mu, mxd, mdt, my, stream);
}